// Gemma4PatchEmbed_39298950758871
// MI455X (gfx1250) — compile-verified
//
#include <hip/hip_runtime.h>

typedef float v2f __attribute__((ext_vector_type(2)));
typedef float v4f __attribute__((ext_vector_type(4)));
typedef float v8f __attribute__((ext_vector_type(8)));

#define PATCH   16
#define CH      3
#define DIM     768          // embed dim (GEMM N)
#define KDIM    768          // C*P*P     (GEMM K)
#define IMG     512
#define NPB     1024         // patches per image
#define BATCH   16
#define MTOT    (BATCH * NPB)   // 16384 (GEMM M)
#define VPOS    10240

#define BM      128
#define BN      128
#define KT      16
#define LDSTR   18           // KT + 2 pad: float2 reads stay 8B aligned, banks spread
#define NSTAGES (KDIM / KT)  // 48

__global__ __launch_bounds__(256)
void gemma4_patch_embed_wmma(const float* __restrict__ pix,
                             const float* __restrict__ w_proj,
                             const float* __restrict__ pos_table,
                             const int*   __restrict__ pos_ids,
                             const unsigned char* __restrict__ pad_pos,
                             float* __restrict__ out)
{
    __shared__ float lds_a[2][BM * LDSTR];
    __shared__ float lds_b[2][BN * LDSTR];

    const int tid    = threadIdx.x;
    const int lane   = tid & 31;
    const int wave   = tid >> 5;     // 8 waves
    const int wave_m = wave & 3;     // 4 waves along M (32 rows each)
    const int wave_n = wave >> 2;    // 2 waves along N (64 cols each)

    const int block_m = blockIdx.x * BM;
    const int block_n = blockIdx.y * BN;

    // ---- global->LDS mapping: each thread moves 8 contiguous floats of A and of B
    const int pr   = tid >> 1;       // 0..127: tile row (A) / tile col (B)
    const int px0  = (tid & 1) * 8;  // half-row offset

    // A row -> pixel coordinates (patch extraction fused into the load)
    const int gm_ld = block_m + pr;
    const int b_img = gm_ld >> 10;
    const int n_pat = gm_ld & 1023;
    const int ph_i  = n_pat >> 5;
    const int pw_i  = n_pat & 31;
    // addr = ((b*3 + c)*512 + ph_i*16 + py)*512 + pw_i*16 + px
    const int a_base = (b_img * 3) * (IMG * IMG) + ph_i * PATCH * IMG + pw_i * PATCH + px0;
    const int b_base = (block_n + pr) * KDIM + px0;

    v8f zero = {0.f, 0.f, 0.f, 0.f, 0.f, 0.f, 0.f, 0.f};
    v8f acc[2][4];
    #pragma unroll
    for (int mi = 0; mi < 2; ++mi)
        #pragma unroll
        for (int ni = 0; ni < 4; ++ni)
            acc[mi][ni] = zero;

    // WMMA fragment addressing (ISA 16x16x4 f32 layout)
    const int lhalf = lane >> 4;     // lanes 16-31 hold K+2,K+3
    const int l16   = lane & 15;
    const int a_row = wave_m * 32;
    const int b_row = wave_n * 64;

    float a_reg[8], b_reg[8];

    auto load_stage = [&](int kt) {
        const int k0 = kt * KT;
        const int c  = k0 >> 8;            // feature -> channel
        const int py = (k0 & 255) >> 4;    // feature -> row in patch
        const float* ap = pix + a_base + (c * IMG + py) * IMG;
        v4f a0 = *(const v4f*)(ap);
        v4f a1 = *(const v4f*)(ap + 4);
        const float* bp = w_proj + b_base + k0;
        v4f b0 = *(const v4f*)(bp);
        v4f b1 = *(const v4f*)(bp + 4);
        #pragma unroll
        for (int q = 0; q < 4; ++q) {
            a_reg[q]     = a0[q] * 2.0f - 1.0f;   // fused normalize 2*(x-0.5)
            a_reg[4 + q] = a1[q] * 2.0f - 1.0f;
            b_reg[q]     = b0[q];
            b_reg[4 + q] = b1[q];
        }
    };
    auto store_stage = [&](int buf) {
        float* la = &lds_a[buf][pr * LDSTR + px0];
        float* lb = &lds_b[buf][pr * LDSTR + px0];
        #pragma unroll
        for (int q = 0; q < 8; ++q) la[q] = a_reg[q];
        #pragma unroll
        for (int q = 0; q < 8; ++q) lb[q] = b_reg[q];
    };

    load_stage(0);
    store_stage(0);
    __syncthreads();

    int buf = 0;
    for (int kt = 0; kt < NSTAGES; ++kt) {
        if (kt + 1 < NSTAGES) {
            load_stage(kt + 1);                       // pipeline next stage into regs
            if (kt + 2 < NSTAGES)                     // gfx1250 global_prefetch_b8
                __builtin_prefetch(w_proj + b_base + (kt + 2) * KT, 0, 1);
        }

        #pragma unroll
        for (int ks = 0; ks < KT; ks += 4) {
            const int kk = ks + 2 * lhalf;
            v2f afrag[2], bfrag[4];
            #pragma unroll
            for (int mi = 0; mi < 2; ++mi)
                afrag[mi] = *(const v2f*)&lds_a[buf][(a_row + mi * 16 + l16) * LDSTR + kk];
            #pragma unroll
            for (int ni = 0; ni < 4; ++ni)
                bfrag[ni] = *(const v2f*)&lds_b[buf][(b_row + ni * 16 + l16) * LDSTR + kk];
            #pragma unroll
            for (int mi = 0; mi < 2; ++mi)
                #pragma unroll
                for (int ni = 0; ni < 4; ++ni)
                    acc[mi][ni] = __builtin_amdgcn_wmma_f32_16x16x4_f32(
                        false, afrag[mi], false, bfrag[ni],
                        (short)0, acc[mi][ni], false, false);
        }

        if (kt + 1 < NSTAGES)
            store_stage(buf ^ 1);    // other buffer: safe to overlap with compute
        __syncthreads();
        buf ^= 1;
    }

    // ---- epilogue: fused position-embedding gather-add + store
    const float* pos0 = pos_table;
    const float* pos1 = pos_table + VPOS * DIM;

    #pragma unroll
    for (int mi = 0; mi < 2; ++mi) {
        #pragma unroll
        for (int r = 0; r < 8; ++r) {
            // C/D layout: VGPR r holds M = r (lanes 0-15) / r+8 (lanes 16-31)
            const int gm = block_m + wave_m * 32 + mi * 16 + r + 8 * lhalf;
            int id0 = pos_ids[gm * 2 + 0];
            int id1 = pos_ids[gm * 2 + 1];
            id0 = id0 < 0 ? 0 : id0;          // jnp.clip(ids, 0)
            id1 = id1 < 0 ? 0 : id1;
            const bool padv = pad_pos[gm] != 0;
            #pragma unroll
            for (int ni = 0; ni < 4; ++ni) {
                const int gd = block_n + wave_n * 64 + ni * 16 + l16;
                float p = 0.0f;
                if (!padv)
                    p = pos0[id0 * DIM + gd] + pos1[id1 * DIM + gd];
                out[gm * DIM + gd] = acc[mi][ni][r] + p;
            }
        }
    }
}

extern "C" void kernel_launch(void* const* d_in, const int* in_sizes, int n_in,
                              void* d_out, int out_size, void* d_ws, size_t ws_size,
                              hipStream_t stream) {
    (void)in_sizes; (void)n_in; (void)d_ws; (void)ws_size; (void)out_size;
    const float*         pix  = (const float*)d_in[0];
    const float*         w    = (const float*)d_in[1];
    const float*         pos  = (const float*)d_in[2];
    const int*           ids  = (const int*)d_in[3];
    const unsigned char* padp = (const unsigned char*)d_in[4];  // numpy bool = 1 byte
    float* out = (float*)d_out;

    dim3 grid(MTOT / BM, DIM / BN);   // 128 x 6 workgroups
    gemma4_patch_embed_wmma<<<grid, 256, 0, stream>>>(pix, w, pos, ids, padp, out);
}